// FAGCNModule_83459804496277
// MI455X (gfx1250) — compile-verified
//
#include <hip/hip_runtime.h>
#include <math.h>

typedef __attribute__((ext_vector_type(2))) float v2f;
typedef __attribute__((ext_vector_type(8))) float v8f;

#define N_NODES 100000
#define N_EDGES 1600000
#define DFEAT   64

// ---------------------------------------------------------------------------
// Kernel 1: zero the output z[N,64] and the degree accumulator deg[N]
// ---------------------------------------------------------------------------
__global__ __launch_bounds__(256)
void k_zero(float* __restrict__ z, float* __restrict__ deg) {
    const int stride = gridDim.x * blockDim.x;
    const int tid = blockIdx.x * blockDim.x + threadIdx.x;
    for (int i = tid; i < N_NODES * DFEAT; i += stride) z[i] = 0.0f;
    for (int i = tid; i < N_NODES; i += stride) deg[i] = 0.0f;
}

// ---------------------------------------------------------------------------
// Kernel 2: in-degree via f32 atomics (L2-resident, 400 KB accumulator)
// ---------------------------------------------------------------------------
__global__ __launch_bounds__(256)
void k_degree(const long long* __restrict__ dst, float* __restrict__ deg) {
    int e = blockIdx.x * blockDim.x + threadIdx.x;
    if (e < N_EDGES) {
        atomicAdd(&deg[(int)dst[e]], 1.0f);
    }
}

// ---------------------------------------------------------------------------
// Kernel 3: norm[i] = clamp(deg,1)^-0.5 in place
// ---------------------------------------------------------------------------
__global__ __launch_bounds__(256)
void k_norm(float* __restrict__ deg_norm) {
    int i = blockIdx.x * blockDim.x + threadIdx.x;
    if (i < N_NODES) {
        float d = fmaxf(deg_norm[i], 1.0f);
        deg_norm[i] = rsqrtf(d);
    }
}

// ---------------------------------------------------------------------------
// Kernel 4: per-node gate dot products via V_WMMA_F32_16X16X4_F32.
//   adot[i] = x[i,:] . w[0:64]     (gate weight on the dst half)
//   bdot[i] = x[i,:] . w[64:128]   (gate weight on the src half)
// One wave handles a 16-node tile: A = X_tile[16 x 64] fed K=4 at a time,
// B = W[64 x 16] with only columns 0 (w_dst) and 1 (w_src) non-zero.
// N_NODES % 16 == 0, so every wave's tile is full and EXEC stays all-ones.
// ---------------------------------------------------------------------------
__global__ __launch_bounds__(256)
void k_node_gate(const float* __restrict__ x, const float* __restrict__ gw,
                 float* __restrict__ adot, float* __restrict__ bdot) {
    const int lane  = threadIdx.x & 31;
    const int wave  = blockIdx.x * (blockDim.x >> 5) + (threadIdx.x >> 5);
    const int base  = wave * 16;                 // first node of this wave's tile
    if (base >= N_NODES) return;                 // wave-uniform exit

    const int row   = lane & 15;                 // A: matrix row / B: column
    const int khalf = (lane >> 4) * 2;           // lanes 0-15 -> K{0,1}, 16-31 -> K{2,3}
    const int node  = base + row;

    // B column selector: col 0 reads w[0:64], col 1 reads w[64:128], rest zero.
    const int   wsel  = (row == 0) ? 0 : 64;     // safe in-bounds address for all lanes
    const float wmask = (row < 2) ? 1.0f : 0.0f;

    const float* __restrict__ xr = x + (size_t)node * DFEAT;

    v8f c = {};                                  // 16x16 f32 accumulator (8 VGPRs)
    #pragma unroll
    for (int ks = 0; ks < 16; ++ks) {            // K = 64 in steps of 4
        const int k0 = ks * 4 + khalf;
        v2f a, b;
        a.x = xr[k0];
        a.y = xr[k0 + 1];
        b.x = gw[wsel + k0]     * wmask;
        b.y = gw[wsel + k0 + 1] * wmask;
        // 8 args: (neg_a, A, neg_b, B, c_mod, C, reuse_a, reuse_b)
        c = __builtin_amdgcn_wmma_f32_16x16x4_f32(
                false, a, false, b, (short)0, c, false, false);
    }

    // C/D layout: element (m,n): m<8 -> VGPR m, lane n ; m>=8 -> VGPR m-8, lane n+16.
    // Column 0 (adot) lives in lanes 0 and 16; column 1 (bdot) in lanes 1 and 17.
    const int mbase = (lane >= 16) ? 8 : 0;
    if (row == 0) {
        #pragma unroll
        for (int j = 0; j < 8; ++j) adot[base + mbase + j] = c[j];
    } else if (row == 1) {
        #pragma unroll
        for (int j = 0; j < 8; ++j) bdot[base + mbase + j] = c[j];
    }
}

// ---------------------------------------------------------------------------
// Kernel 5: edge scatter. One wave per edge; lane l owns features {2l, 2l+1}.
//   g    = tanh(adot[dst] + bdot[src] + b)
//   coef = g * norm[dst] * norm[src]
//   z[dst,:] += coef * x[src,:]     (f32 atomics, resolved in L2)
// ---------------------------------------------------------------------------
__global__ __launch_bounds__(256)
void k_edge(const float* __restrict__ x,
            const long long* __restrict__ src, const long long* __restrict__ dst,
            const float* __restrict__ adot, const float* __restrict__ bdot,
            const float* __restrict__ norm, const float* __restrict__ gb,
            float* __restrict__ z) {
    const int wave = blockIdx.x * (blockDim.x >> 5) + (threadIdx.x >> 5);
    const int lane = threadIdx.x & 31;
    if (wave >= N_EDGES) return;

    const int s = (int)src[wave];
    const int d = (int)dst[wave];

    const float g    = tanhf(adot[d] + bdot[s] + gb[0]);
    const float coef = g * norm[d] * norm[s];

    const float2 v = ((const float2*)(x + (size_t)s * DFEAT))[lane];
    float* zp = z + (size_t)d * DFEAT + 2 * lane;
    atomicAdd(zp,     coef * v.x);
    atomicAdd(zp + 1, coef * v.y);
}

// ---------------------------------------------------------------------------
// Host-side launcher
// ---------------------------------------------------------------------------
extern "C" void kernel_launch(void* const* d_in, const int* in_sizes, int n_in,
                              void* d_out, int out_size, void* d_ws, size_t ws_size,
                              hipStream_t stream) {
    const float*     x   = (const float*)d_in[0];      // [N, 64]
    const float*     gw  = (const float*)d_in[1];      // [1, 128]
    const float*     gb  = (const float*)d_in[2];      // [1]
    const long long* src = (const long long*)d_in[3];  // [E] int64
    const long long* dst = (const long long*)d_in[4];  // [E] int64
    float* z = (float*)d_out;                          // [N, 64]

    // workspace: norm | adot | bdot  (3 * N floats = 1.2 MB)
    float* norm = (float*)d_ws;
    float* adot = norm + N_NODES;
    float* bdot = adot + N_NODES;

    (void)in_sizes; (void)n_in; (void)out_size; (void)ws_size;

    // 1) zero z and deg
    k_zero<<<2048, 256, 0, stream>>>(z, norm);

    // 2) in-degree
    k_degree<<<(N_EDGES + 255) / 256, 256, 0, stream>>>(dst, norm);

    // 3) norm = clamp(deg,1)^-0.5
    k_norm<<<(N_NODES + 255) / 256, 256, 0, stream>>>(norm);

    // 4) WMMA node-gate dots: 6250 waves of 16 nodes each
    const int gate_waves   = N_NODES / 16;             // 6250
    const int gate_threads = gate_waves * 32;
    k_node_gate<<<(gate_threads + 255) / 256, 256, 0, stream>>>(x, gw, adot, bdot);

    // 5) edge scatter: one wave per edge, 8 edges per 256-thread block
    k_edge<<<N_EDGES / 8, 256, 0, stream>>>(x, src, dst, adot, bdot, norm, gb, z);
}